// PhysicsInformedLoss_40501541601298
// MI455X (gfx1250) — compile-verified
//
#include <hip/hip_runtime.h>
#include <hip/hip_bf16.h>

// ---------------------------------------------------------------------------
// PhysicsInformedLoss for MI455X (gfx1250, wave32)
// Memory-bound stencil+reduction: stage tiles to LDS via CDNA5 async
// global->LDS loads (ASYNCcnt), compute 3x3 stencils from LDS, block
// reduction via V_WMMA_F32_16X16X4_F32 (A = ones -> column-sum accumulate),
// atomic per-image accumulators, tiny finish kernels.
// ---------------------------------------------------------------------------

#define AS1 __attribute__((address_space(1)))
#define AS3 __attribute__((address_space(3)))

typedef int   pi_int4 __attribute__((ext_vector_type(4)));
typedef float pi_v2f  __attribute__((ext_vector_type(2)));
typedef float pi_v8f  __attribute__((ext_vector_type(8)));

#if defined(__has_builtin)
#  if __has_builtin(__builtin_amdgcn_global_load_async_to_lds_b128)
#    define PI_USE_ASYNC 1
#  endif
#  if __has_builtin(__builtin_amdgcn_wmma_f32_16x16x4_f32)
#    define PI_USE_WMMA 1
#  endif
#endif
#ifndef PI_USE_ASYNC
#  define PI_USE_ASYNC 0
#endif
#ifndef PI_USE_WMMA
#  define PI_USE_WMMA 0
#endif

namespace {
constexpr int   B      = 128;
constexpr int   H      = 256;
constexpr int   W      = 256;
constexpr int   PLANE  = H * W;          // 65536
constexpr int   TR     = 16;             // output rows per tile
constexpr int   HR     = TR + 2;         // halo rows in LDS
constexpr int   CH_LDS = HR * W;         // 4608 floats / channel
constexpr float DXc    = 0.01f;
constexpr float INV_DX = 1.0f / DXc;     // 100
constexpr float KVD    = 0.001f / 0.01f; // VISC/DX = 0.1
constexpr float U0c    = 1.0f;
constexpr float L_CON  = 0.3f;
constexpr float L_BC   = 0.2f;
constexpr float L_POI  = 0.5f;           // 1 - 0.3 - 0.2
}

__device__ __forceinline__ void pi_wait_async0() {
#if PI_USE_ASYNC
#  if defined(__has_builtin) && __has_builtin(__builtin_amdgcn_s_wait_asynccnt)
    __builtin_amdgcn_s_wait_asynccnt(0);
#  else
    asm volatile("s_wait_asynccnt 0" ::: "memory");
#  endif
#endif
}

__device__ __forceinline__ float waveSum(float v) {
    // wave32 butterfly reduction
    #pragma unroll
    for (int m = 16; m > 0; m >>= 1) v += __shfl_xor(v, m, 32);
    return v;
}

// ws layout (floats): [0,128)  sum|cont|   per image
//                     [128,256) sum|dudt|  per image
//                     [256,384) sum|dvdt|  per image
//                     [384,512) bc_b = |y0|+|yL|+|x0|+|xL| per image
__global__ __launch_bounds__(256) void pi_zero(float* __restrict__ ws) {
    int t = blockIdx.x * blockDim.x + threadIdx.x;
    if (t < 512) ws[t] = 0.0f;
}

__global__ __launch_bounds__(256) void pi_interior(const float* __restrict__ g,
                                                   float* __restrict__ ws) {
    __shared__ float tile[3 * CH_LDS];   // 55,296 B
    const int b   = blockIdx.y;
    const int r0  = blockIdx.x * TR;
    const int tid = threadIdx.x;
    const float* img = g + (size_t)b * 3 * PLANE;

    // ---- stage (TR+2) x 256 x {u,v,p} into LDS, 16B per lane per issue ----
    constexpr int NF4 = 3 * HR * (W / 4);       // 3456 float4 transfers
    for (int f = tid; f < NF4; f += 256) {
        int c   = f / (HR * (W / 4));
        int rem = f - c * (HR * (W / 4));
        int r   = rem >> 6;                     // halo row 0..17
        int q   = rem & 63;                     // float4 within row
        int gr  = r0 + r; if (gr > H - 1) gr = H - 1;
        int gidx = c * PLANE + gr * W + (q << 2);
        int lidx = c * CH_LDS + r * W + (q << 2);
#if PI_USE_ASYNC
        __builtin_amdgcn_global_load_async_to_lds_b128(
            (AS1 pi_int4*)(img + gidx), (AS3 pi_int4*)(&tile[lidx]), 0, 0);
#else
        *(float4*)(&tile[lidx]) = *(const float4*)(img + gidx);
#endif
    }
    pi_wait_async0();
    __syncthreads();

#define U_(r, c) tile[(r) * W + (c)]
#define V_(r, c) tile[CH_LDS + (r) * W + (c)]
#define P_(r, c) tile[2 * CH_LDS + (r) * W + (c)]

    const int j = tid;            // column
    float a_cont = 0.0f, a_du = 0.0f, a_dv = 0.0f;

    #pragma unroll 4
    for (int li = 0; li < TR; ++li) {
        const int i = r0 + li;                 // output row
        if (i > 253) break;                    // uniform across block

        if (j <= 253) {                        // continuity at (i, j)
            float r = (U_(li + 1, j + 1) - U_(li + 1, j)
                     + V_(li + 1, j + 1) - V_(li,     j + 1)) * INV_DX;
            a_cont += fabsf(r);
        }

        if (j <= 252) {                        // dudt at (i, j)
            float u10 = U_(li + 1, j);
            float u11 = U_(li + 1, j + 1);
            float u12 = U_(li + 1, j + 2);
            float u01 = U_(li,     j + 1);
            float u21 = U_(li + 2, j + 1);
            float v01 = V_(li,     j + 1);
            float v02 = V_(li,     j + 2);
            float v11 = V_(li + 1, j + 1);
            float v12 = V_(li + 1, j + 2);
            float p11 = P_(li + 1, j + 1);
            float p12 = P_(li + 1, j + 2);

            float uea = 0.5f * (u10 + u11);                    // fe(i+1, j)
            float ueb = 0.5f * (u11 + u12);                    // fe(i+1, j+1)
            float fea = uea * uea - KVD * (u11 - u10);
            float feb = ueb * ueb - KVD * (u12 - u11);

            float vna = 0.5f * (v01 + v02);                    // fn(i,   j+1)
            float aya = 0.5f * (u01 + u11);
            float fna = vna * aya - KVD * (u11 - u01);
            float vnb = 0.5f * (v11 + v12);                    // fn(i+1, j+1)
            float ayb = 0.5f * (u11 + u21);
            float fnb = vnb * ayb - KVD * (u21 - u11);

            float dudt = -((feb - fea) + (fnb - fna) + (p12 - p11)) * INV_DX;
            a_du += fabsf(dudt);
        }

        if (j <= 253 && i <= 252) {            // dvdt at (i, j)
            float v10 = V_(li + 1, j);
            float v11 = V_(li + 1, j + 1);
            float v12 = V_(li + 1, j + 2);
            float v01 = V_(li,     j + 1);
            float v21 = V_(li + 2, j + 1);
            float u10 = U_(li + 1, j);
            float u20 = U_(li + 2, j);
            float u11 = U_(li + 1, j + 1);
            float u21 = U_(li + 2, j + 1);
            float p11 = P_(li + 1, j + 1);
            float p21 = P_(li + 2, j + 1);

            float ayua = 0.5f * (u10 + u20);                   // fe(i+1, j)
            float axva = 0.5f * (v10 + v11);
            float fea  = ayua * axva - KVD * (v11 - v10);
            float ayub = 0.5f * (u11 + u21);                   // fe(i+1, j+1)
            float axvb = 0.5f * (v11 + v12);
            float feb  = ayub * axvb - KVD * (v12 - v11);

            float ayva = 0.5f * (v01 + v11);                   // fn(i,   j+1)
            float fna  = ayva * ayva - KVD * (v11 - v01);
            float ayvb = 0.5f * (v11 + v21);                   // fn(i+1, j+1)
            float fnb  = ayvb * ayvb - KVD * (v21 - v11);

            float dvdt = -((feb - fea) + (fnb - fna) + (p21 - p11)) * INV_DX;
            a_dv += fabsf(dvdt);
        }
    }
#undef U_
#undef V_
#undef P_

    // ---- block reduction: 3 x 256 partials -> 3 sums ----
    __syncthreads();                     // everyone done reading tile
    float* red = tile;                   // reuse LDS scratch (768 floats)
    red[tid]       = a_cont;
    red[256 + tid] = a_du;
    red[512 + tid] = a_dv;
    __syncthreads();

    if (tid < 32) {                      // wave 0 only; EXEC all-ones in wave
#if PI_USE_WMMA
        // D = ones(16x4) x B(4x16) + C  => every D row = column sums of B.
        // Total of all 64 B values = sum_n D[0,n]; exact for any bijective
        // lane->(k,n) operand layout, so this is layout-robust.
        const pi_v2f ones = {1.0f, 1.0f};
        float sums[3];
        #pragma unroll
        for (int q = 0; q < 3; ++q) {
            pi_v8f c = {};
            #pragma unroll
            for (int i = 0; i < 4; ++i) {   // 4 x 64 = 256 partials
                pi_v2f bm;
                bm.x = red[q * 256 + i * 64 + tid];
                bm.y = red[q * 256 + i * 64 + 32 + tid];
                c = __builtin_amdgcn_wmma_f32_16x16x4_f32(
                        false, ones, false, bm, (short)0, c, false, false);
            }
            float s = c[0];                 // lane n<16: column sum n (M=0 row)
            #pragma unroll
            for (int m = 8; m > 0; m >>= 1) s += __shfl_xor(s, m, 16);
            sums[q] = s;                    // lane 0: total
        }
        if (tid == 0) {
            atomicAdd(&ws[b],         sums[0]);
            atomicAdd(&ws[B + b],     sums[1]);
            atomicAdd(&ws[2 * B + b], sums[2]);
        }
#else
        float sums[3];
        #pragma unroll
        for (int q = 0; q < 3; ++q) {
            float s = 0.0f;
            #pragma unroll
            for (int w = 0; w < 8; ++w) s += red[q * 256 + w * 32 + tid];
            sums[q] = waveSum(s);
        }
        if (tid == 0) {
            atomicAdd(&ws[b],         sums[0]);
            atomicAdd(&ws[B + b],     sums[1]);
            atomicAdd(&ws[2 * B + b], sums[2]);
        }
#endif
    }
}

__global__ __launch_bounds__(256) void pi_boundary(const float* __restrict__ g,
                                                   float* __restrict__ ws) {
    const int b = blockIdx.x;
    const int t = threadIdx.x;
    const float* u = g + (size_t)b * 3 * PLANE;
    const float* v = u + PLANE;
    const float* p = v + PLANE;

    float y0 = 0.f, yL = 0.f, x0 = 0.f, xL = 0.f;
    if (t >= 1 && t <= 253) {                       // N-1 = 253 entries
        y0 += u[0 * W + t] + u[1 * W + t];
        yL += 2.0f * U0c - (u[254 * W + t] + u[255 * W + t]);
        x0 += v[t * W + 0] + v[t * W + 1];
        xL += v[t * W + 255] + v[t * W + 254];
    }
    if (t >= 1 && t <= 254) {                       // N = 254 entries
        y0 += v[0 * W + t] + p[0 * W + t];
        yL += v[255 * W + t] + p[255 * W + t];
        x0 += u[t * W + 0] + p[t * W + 0];
        xL += u[t * W + 255] + p[t * W + 255];
    }

    __shared__ float red[4][8];
    float vals[4] = {y0, yL, x0, xL};
    #pragma unroll
    for (int q = 0; q < 4; ++q) {
        float s = waveSum(vals[q]);
        if ((t & 31) == 0) red[q][t >> 5] = s;
    }
    __syncthreads();
    if (t == 0) {
        float s[4] = {0.f, 0.f, 0.f, 0.f};
        #pragma unroll
        for (int q = 0; q < 4; ++q)
            for (int w = 0; w < 8; ++w) s[q] += red[q][w];
        ws[3 * B + b] = fabsf(s[0]) + fabsf(s[1]) + fabsf(s[2]) + fabsf(s[3]);
    }
}

__global__ __launch_bounds__(128) void pi_final(const float* __restrict__ ws,
                                                float* __restrict__ out) {
    const int t = threadIdx.x;                      // 128 threads, one per image
    float c  = ws[t]         * (1.0f / (254.0f * 254.0f));
    float pu = ws[B + t]     * (1.0f / (254.0f * 253.0f));
    float pv = ws[2 * B + t] * (1.0f / (253.0f * 254.0f));
    float bc = ws[3 * B + t];
    // mean over batch of (L_CON*cont_b + L_POI*poisson_b) + L_BC * sum_b bc_b
    float val = (L_CON * c + L_POI * (pu + pv)) * (1.0f / (float)B) + L_BC * bc;

    __shared__ float red[4];
    float s = waveSum(val);
    if ((t & 31) == 0) red[t >> 5] = s;
    __syncthreads();
    if (t == 0) out[0] = red[0] + red[1] + red[2] + red[3];
}

extern "C" void kernel_launch(void* const* d_in, const int* in_sizes, int n_in,
                              void* d_out, int out_size, void* d_ws, size_t ws_size,
                              hipStream_t stream) {
    const float* g = (const float*)d_in[0];      // (128, 3, 256, 256) f32
    float* out = (float*)d_out;                  // 1 float
    float* ws  = (float*)d_ws;                   // uses 512 floats
    (void)in_sizes; (void)n_in; (void)out_size; (void)ws_size;

    pi_zero<<<2, 256, 0, stream>>>(ws);
    dim3 grid(16, 128);                          // 16 row-tiles x 128 images
    pi_interior<<<grid, 256, 0, stream>>>(g, ws);
    pi_boundary<<<128, 256, 0, stream>>>(g, ws);
    pi_final<<<1, 128, 0, stream>>>(ws, out);
}